// DilatedGraphConvolutionCell_79439715106955
// MI455X (gfx1250) — compile-verified
//
#include <hip/hip_runtime.h>
#include <math.h>
#include <stdint.h>

// ---------------------------------------------------------------------------
// CDNA5 (gfx1250) fp32 WMMA helpers.  V_WMMA_F32_16X16X4_F32:
//   A: 16x4 f32  -> 2 VGPRs/lane (lane&15 = M row; lane>=16 => K=2,3)
//   B:  4x16 f32 -> 2 VGPRs/lane (lane&15 = N col; lane>=16 => K=2,3)
//   C/D: 16x16 f32 -> 8 VGPRs/lane (lane&15 = N col; vgpr i => M = i + 8*(lane>>4))
// ---------------------------------------------------------------------------
typedef float v2f __attribute__((ext_vector_type(2)));
typedef float v8f __attribute__((ext_vector_type(8)));
typedef unsigned u32x4 __attribute__((ext_vector_type(4)));
typedef int i32x8 __attribute__((ext_vector_type(8)));
typedef int i32x4 __attribute__((ext_vector_type(4)));

static __device__ __forceinline__ v8f wmma_f32(v2f a, v2f b, v8f c) {
  return __builtin_amdgcn_wmma_f32_16x16x4_f32(false, a, false, b, (short)0, c,
                                               false, false);
}

#define N_NODES 1024
#define FDIM    64
#define LDIM    64
#define NF      65536   // N_NODES * FDIM

// ---------------------------------------------------------------------------
// Tensor Data Mover: async-load a 2-D fp32 tile (rows x width, row stride in
// elements) from global memory into LDS.  D# packing per CDNA5 ISA ch.8:
//   group0: [1:0]=count=1 | lds_addr | global_addr[56:0] | type=2 (bits 127:126)
//   group1: data_size=2 (4B) | tensor_dim0/1 | tile_dim0/1 (tile_dim2=0)
//           | tensor_dim0_stride (48b); no mask/pad/iterate/barrier.
//   remaining groups: zero (2-D tile; higher-dim descriptors disabled).
//   Tracked by TENSORcnt.  This toolchain exposes the 6-arg builtin:
//   (u32x4, i32x8, i32x4, i32x4, i32x8, i32 cpol).
// ---------------------------------------------------------------------------
static __device__ __forceinline__ void tdm_load_2d_f32(
    unsigned lds_off, const float* gptr, unsigned width, unsigned rows,
    unsigned long row_stride_elems)
{
  const unsigned long ga = (unsigned long)(uintptr_t)gptr;
  u32x4 g0;
  g0[0] = 1u;                                            // count=1, user mode
  g0[1] = lds_off;                                       // LDS byte address
  g0[2] = (unsigned)(ga & 0xffffffffu);                  // global_addr[31:0]
  g0[3] = (unsigned)((ga >> 32) & 0x01ffffffu) | (2u << 30); // [56:32] | type=2

  i32x8 g1;
  g1[0] = (int)(2u << 16);                               // data_size=4B
  g1[1] = (int)(width << 16);                            // tensor_dim0[15:0]
  g1[2] = (int)((width >> 16) | (rows << 16));           // tdim0 hi | tdim1 lo
  g1[3] = (int)((rows >> 16) | (width << 16));           // tdim1 hi | tile_dim0
  g1[4] = (int)(rows & 0xffffu);                         // tile_dim1 (tile_dim2=0)
  g1[5] = (int)(row_stride_elems & 0xffffffffu);         // dim0_stride[31:0]
  g1[6] = (int)((row_stride_elems >> 32) & 0xffffu);     // dim0_stride[47:32]
  g1[7] = 0;

  const i32x4 z4 = {0, 0, 0, 0};
  const i32x8 z8 = {0, 0, 0, 0, 0, 0, 0, 0};
  __builtin_amdgcn_tensor_load_to_lds(g0, g1, z4, z4, z8, 0);
}

// ---------------------------------------------------------------------------
// Generic fp32 WMMA GEMM:  C[M,N] (beta? += : =) A[M,K] @ B[K,N] + bias[N]
// B panels (64 K-rows x 16 N-cols) are staged into LDS by the TDM, double
// buffered so the DMA of chunk c+1 overlaps the WMMA loop of chunk c.
// A fragments: contiguous 64-bit loads when ACONTIG (ldac==1), else strided.
// grid = (N/16, M/16), block = 32 (one wave).
// ---------------------------------------------------------------------------
template <bool ACONTIG>
__global__ __launch_bounds__(32) void gemm_wmma_f32(
    const float* __restrict__ A, long ldar, long ldac,
    const float* __restrict__ B, long ldb,
    const float* __restrict__ bias,
    float* __restrict__ C, long ldc,
    int K, int beta)
{
  __shared__ float sB[2][64 * 16];   // double-buffered B panel (8 KB)

  const int lane = threadIdx.x & 31;
  const int r15  = lane & 15;          // M row for A, N col for B/C
  const int hi   = lane >> 4;
  const int koff = hi * 2;
  const long n0 = (long)blockIdx.x * 16;
  const long m0 = (long)blockIdx.y * 16;
  const float* Arow = A + (m0 + r15) * ldar;

  const unsigned lds0 = (unsigned)(uintptr_t)&sB[0][0];
  const unsigned lds1 = (unsigned)(uintptr_t)&sB[1][0];

  const int nchunks = (K + 63) >> 6;
  {
    const unsigned rows0 = (unsigned)(K < 64 ? K : 64);
    tdm_load_2d_f32(lds0, B + n0, 16u, rows0, (unsigned long)ldb);
  }

  v8f acc = {};
  for (int c = 0; c < nchunks; ++c) {
    const int kb = c * 64;
    const int rows = (K - kb) < 64 ? (K - kb) : 64;
    if (c + 1 < nchunks) {
      const int kb2 = kb + 64;
      const unsigned rows2 = (unsigned)((K - kb2) < 64 ? (K - kb2) : 64);
      tdm_load_2d_f32((c & 1) ? lds0 : lds1, B + (long)kb2 * ldb + n0, 16u,
                      rows2, (unsigned long)ldb);
      __builtin_amdgcn_s_wait_tensorcnt(1);  // chunk c resident
    } else {
      __builtin_amdgcn_s_wait_tensorcnt(0);
    }
    const float* bb = sB[c & 1];
#pragma unroll 4
    for (int ks = 0; ks < rows; ks += 4) {
      const int kk = ks + koff;
      v2f a, b;
      if (ACONTIG) {
        a = *(const v2f*)(Arow + kb + kk);
      } else {
        a.x = Arow[(long)(kb + kk) * ldac];
        a.y = Arow[(long)(kb + kk + 1) * ldac];
      }
      b.x = bb[kk * 16 + r15];
      b.y = bb[(kk + 1) * 16 + r15];
      acc = wmma_f32(a, b, acc);
    }
  }

  const long col = n0 + r15;
  const float bv = bias ? bias[col] : 0.0f;
  for (int i = 0; i < 8; ++i) {
    const long row = m0 + i + 8 * hi;
    const long idx = row * ldc + col;
    float v = acc[i] + bv;
    if (beta) v += C[idx];
    C[idx] = v;
  }
}

// ---------------------------------------------------------------------------
// Row-wise LayerNorm + ReLU (in place).  grid = nrows, block = 256.
// ---------------------------------------------------------------------------
__global__ void ln_relu_rows(float* __restrict__ X, const float* __restrict__ g,
                             const float* __restrict__ b, int ncols)
{
  const long rowbase = (long)blockIdx.x * ncols;
  const int tid = threadIdx.x;
  __shared__ float red[256];

  float s1 = 0.f, s2 = 0.f;
  for (int c = tid; c < ncols; c += blockDim.x) {
    float v = X[rowbase + c];
    s1 += v; s2 += v * v;
  }
  red[tid] = s1; __syncthreads();
  for (int s = 128; s > 0; s >>= 1) { if (tid < s) red[tid] += red[tid + s]; __syncthreads(); }
  const float mu = red[0] / ncols;
  __syncthreads();
  red[tid] = s2; __syncthreads();
  for (int s = 128; s > 0; s >>= 1) { if (tid < s) red[tid] += red[tid + s]; __syncthreads(); }
  const float var = red[0] / ncols - mu * mu;
  const float rstd = rsqrtf(var + 1e-5f);
  __syncthreads();
  for (int c = tid; c < ncols; c += blockDim.x) {
    float v = (X[rowbase + c] - mu) * rstd * g[c] + b[c];
    X[rowbase + c] = v > 0.f ? v : 0.f;
  }
}

// ---------------------------------------------------------------------------
// Layout helpers.
// ---------------------------------------------------------------------------
__global__ void xflat_to_x(const float* __restrict__ Xf, float* __restrict__ X)
{
  long i = (long)blockIdx.x * blockDim.x + threadIdx.x;
  const long total = (long)NF * LDIM;
  if (i >= total) return;
  const int l = (int)(i % LDIM);
  const long nf = i / LDIM;
  X[i] = Xf[(long)l * NF + nf];
}

__global__ void z_to_x(const float* __restrict__ Z, float* __restrict__ X, int mcount)
{
  long i = (long)blockIdx.x * blockDim.x + threadIdx.x;
  const long total = (long)NF * mcount;
  if (i >= total) return;
  const int mm = (int)(i % mcount);
  const long nf = i / mcount;
  X[i] = Z[(long)mm * NF + nf];
}

__global__ void zero_f32(float* __restrict__ p, long n)
{
  long i = (long)blockIdx.x * blockDim.x + threadIdx.x;
  if (i < n) p[i] = 0.f;
}

__global__ void copy_f32(const float* __restrict__ s, float* __restrict__ d, long n)
{
  long i = (long)blockIdx.x * blockDim.x + threadIdx.x;
  if (i < n) d[i] = s[i];
}

// ---------------------------------------------------------------------------
// Fused dilated graph conv step for one k of one dilation stage.
//   per (16-node tile, time slice m), per path {fwd(A1,Wf), bwd(A2,Wb)}:
//     T = Ua @ B            (16x64  WMMA -> LDS)
//     S = T @ Ub^T          (16x1024 WMMA -> LDS, thresholded at 0.05)
//     A = softmax_rows(S)   (dn degree terms == softmax row sums == 1)
//     macc = A @ Xk         (16x64  WMMA, A-frags = ds_load_b64 from LDS)
//     oacc += macc @ W[k]
//   Z[m, tile, :] += tanh(oacc + bconv)
// Ua[n,l] = emb[l, n*F + a], Ub[q,p] = emb[p, q*F + b], a/b = gamma*t mod 64.
// grid = (64, mcount), block = 32 (one wave).
// ---------------------------------------------------------------------------
__global__ __launch_bounds__(32) void graph_conv_wmma(
    const float* __restrict__ emb,    // (L, N*F)
    const float* __restrict__ Bmat,   // (64, 64)
    const float* __restrict__ Xin,    // (N, F, L0)
    const float* __restrict__ Wfk,    // (64, 64)  Wf[k]
    const float* __restrict__ Wbk,    // (64, 64)  Wb[k]
    const float* __restrict__ bconv,  // (64)
    float* __restrict__ Z,            // (mcount, N, F)
    int L0, int d, int k, int gamma)
{
  const int lane = threadIdx.x & 31;
  const int r15  = lane & 15;
  const int hi   = lane >> 4;
  const int koff = hi * 2;
  const int nb = blockIdx.x * 16;     // node-tile base
  const int m  = blockIdx.y;          // time slice
  const int t_s = (d - 1) + m * d;
  const int t_k = t_s - k;
  const int a1 = (gamma * t_k) & 63;
  const int b1 = (gamma * t_s) & 63;

  __shared__ float sT[16 * 64];       // T = Ua @ B
  __shared__ float sM[16 * 64];       // msg staging
  __shared__ float sS[16 * 1024];     // S row-block / softmax A

  v8f oacc[4] = {};

  for (int path = 0; path < 2; ++path) {
    const int afeat = (path == 0) ? a1 : b1;
    const int bfeat = (path == 0) ? b1 : a1;
    const float* W  = (path == 0) ? Wfk : Wbk;

    // ---- T = Ua @ Bmat (16x64), K = 64 over l ----
    for (int ct = 0; ct < 4; ++ct) {
      const int p0 = ct * 16;
      v8f tacc = {};
#pragma unroll 4
      for (int kb = 0; kb < 64; kb += 4) {
        v2f a, b;
        a.x = emb[(long)(kb + koff)     * NF + (long)(nb + r15) * FDIM + afeat];
        a.y = emb[(long)(kb + koff + 1) * NF + (long)(nb + r15) * FDIM + afeat];
        b.x = Bmat[(kb + koff)     * 64 + p0 + r15];
        b.y = Bmat[(kb + koff + 1) * 64 + p0 + r15];
        tacc = wmma_f32(a, b, tacc);
      }
      for (int i = 0; i < 8; ++i) sT[(i + 8 * hi) * 64 + p0 + r15] = tacc[i];
    }
    __syncthreads();

    // ---- S row-block (16 x 1024) = T @ Ub^T, thresholded at DELTA ----
    for (int qt = 0; qt < 64; ++qt) {
      const int q0 = qt * 16;
      v8f sacc = {};
#pragma unroll 4
      for (int kb = 0; kb < 64; kb += 4) {
        v2f a = *(const v2f*)&sT[r15 * 64 + kb + koff];   // ds_load_b64
        v2f b;
        b.x = emb[(long)(kb + koff)     * NF + (long)(q0 + r15) * FDIM + bfeat];
        b.y = emb[(long)(kb + koff + 1) * NF + (long)(q0 + r15) * FDIM + bfeat];
        sacc = wmma_f32(a, b, sacc);
      }
      for (int i = 0; i < 8; ++i) {
        float v = sacc[i];
        v = (v >= 0.05f) ? v : 0.f;
        sS[(i + 8 * hi) * 1024 + q0 + r15] = v;
      }
    }
    __syncthreads();

    // ---- row-wise softmax over 1024 entries (wave32 shfl reductions) ----
    for (int r = 0; r < 16; ++r) {
      float mx = -3.4e38f;
      for (int c = lane; c < 1024; c += 32) mx = fmaxf(mx, sS[r * 1024 + c]);
      for (int off = 16; off > 0; off >>= 1) mx = fmaxf(mx, __shfl_xor(mx, off, 32));
      float sum = 0.f;
      for (int c = lane; c < 1024; c += 32) {
        const float e = expf(sS[r * 1024 + c] - mx);
        sS[r * 1024 + c] = e;
        sum += e;
      }
      for (int off = 16; off > 0; off >>= 1) sum += __shfl_xor(sum, off, 32);
      const float inv = 1.f / sum;
      for (int c = lane; c < 1024; c += 32) sS[r * 1024 + c] *= inv;
    }
    __syncthreads();

    // ---- macc = A @ Xk (16x64), K = 1024 over q ----
    v8f macc[4] = {};
    for (int qt = 0; qt < 64; ++qt) {
      const int q0 = qt * 16;
#pragma unroll
      for (int ks = 0; ks < 16; ks += 4) {
        const v2f a = *(const v2f*)&sS[r15 * 1024 + q0 + ks + koff];  // b64
        const long q = q0 + ks + koff;
#pragma unroll
        for (int ft = 0; ft < 4; ++ft) {
          const int f0 = ft * 16;
          v2f b;
          b.x = Xin[(q)     * (long)FDIM * L0 + (long)(f0 + r15) * L0 + t_k];
          b.y = Xin[(q + 1) * (long)FDIM * L0 + (long)(f0 + r15) * L0 + t_k];
          macc[ft] = wmma_f32(a, b, macc[ft]);
        }
      }
    }
    __syncthreads();
    for (int ft = 0; ft < 4; ++ft)
      for (int i = 0; i < 8; ++i)
        sM[(i + 8 * hi) * 64 + ft * 16 + r15] = macc[ft][i];
    __syncthreads();

    // ---- oacc += macc @ W[k] (16x64), K = 64 ----
#pragma unroll 4
    for (int kb = 0; kb < 64; kb += 4) {
      const v2f a = *(const v2f*)&sM[r15 * 64 + kb + koff];           // b64
#pragma unroll
      for (int ft = 0; ft < 4; ++ft) {
        v2f b;
        b.x = W[(kb + koff)     * 64 + ft * 16 + r15];
        b.y = W[(kb + koff + 1) * 64 + ft * 16 + r15];
        oacc[ft] = wmma_f32(a, b, oacc[ft]);
      }
    }
    __syncthreads();
  }

  // ---- Z[m, nb+.., f] += tanh(oacc + bconv[f]) ----
  for (int ft = 0; ft < 4; ++ft) {
    const int f = ft * 16 + r15;
    const float bc = bconv[f];
    for (int i = 0; i < 8; ++i) {
      const long idx = (long)m * NF + (long)(nb + i + 8 * hi) * FDIM + f;
      Z[idx] += tanhf(oacc[ft][i] + bc);
    }
  }
}

// ---------------------------------------------------------------------------
// Host-side orchestration.
// ---------------------------------------------------------------------------
extern "C" void kernel_launch(void* const* d_in, const int* in_sizes, int n_in,
                              void* d_out, int out_size, void* d_ws, size_t ws_size,
                              hipStream_t stream)
{
  const float* observation   = (const float*)d_in[0];   // (N, NDF, L)
  const float* time_features = (const float*)d_in[1];   // (L, NTF=8)
  const float* layer_initial = (const float*)d_in[2];   // (L, N)
  const float* sW1  = (const float*)d_in[3];
  const float* sb1  = (const float*)d_in[4];
  const float* slng = (const float*)d_in[5];
  const float* slnb = (const float*)d_in[6];
  const float* sW2  = (const float*)d_in[7];
  const float* sb2  = (const float*)d_in[8];
  const float* tW1  = (const float*)d_in[9];
  const float* tb1  = (const float*)d_in[10];
  const float* tlng = (const float*)d_in[11];
  const float* tlnb = (const float*)d_in[12];
  const float* tW2  = (const float*)d_in[13];
  const float* tb2  = (const float*)d_in[14];
  const float* Bmat = (const float*)d_in[15];
  const float* fW1  = (const float*)d_in[16];
  const float* fb1  = (const float*)d_in[17];
  const float* f1g  = (const float*)d_in[18];
  const float* f1b  = (const float*)d_in[19];
  const float* fW2  = (const float*)d_in[20];
  const float* fb2  = (const float*)d_in[21];
  const float* f2g  = (const float*)d_in[22];
  const float* f2b  = (const float*)d_in[23];
  const float* fW3  = (const float*)d_in[24];
  const float* fb3  = (const float*)d_in[25];
  const float* Wf   = (const float*)d_in[26];  // (2, 64, 64)
  const float* Wb   = (const float*)d_in[27];  // (2, 64, 64)
  const float* bconv= (const float*)d_in[28];
  float* out = (float*)d_out;                  // (2, N, F) = 131072 floats

  float* ws    = (float*)d_ws;
  float* emb   = ws;                  // 4,194,304
  float* Xflat = emb   + 4194304;     // 4,194,304
  float* X0    = Xflat + 4194304;     // 4,194,304  (N, F, 64)
  float* X1    = X0    + 4194304;     // 2,097,152  (N, F, 32)
  float* Z     = X1    + 2097152;     // 2,097,152  (max 32, N, F)
  float* h_s   = Z     + 2097152;     // 16384
  float* h_t   = h_s   + 16384;       // 16384
  float* hf1   = h_t   + 16384;       // 65536
  float* hf2   = hf1   + 65536;       // 32768

  const dim3 w(32, 1, 1);

  // ---- spatial embedding branch: h_s = relu(ln(layer_initial @ sW1 + sb1)) ----
  gemm_wmma_f32<true><<<dim3(16, 4), w, 0, stream>>>(layer_initial, 1024, 1,
                                                     sW1, 256, sb1, h_s, 256,
                                                     1024, 0);
  ln_relu_rows<<<64, 256, 0, stream>>>(h_s, slng, slnb, 256);

  // ---- temporal branch: h_t = relu(ln(time_features @ tW1 + tb1)) ----
  gemm_wmma_f32<true><<<dim3(16, 4), w, 0, stream>>>(time_features, 8, 1,
                                                     tW1, 256, tb1, h_t, 256,
                                                     8, 0);
  ln_relu_rows<<<64, 256, 0, stream>>>(h_t, tlng, tlnb, 256);

  // ---- emb = h_s @ sW2 + sb2 + h_t @ tW2 + tb2   (64 x 65536) ----
  gemm_wmma_f32<true><<<dim3(4096, 4), w, 0, stream>>>(h_s, 256, 1, sW2, NF,
                                                       sb2, emb, NF, 256, 0);
  gemm_wmma_f32<true><<<dim3(4096, 4), w, 0, stream>>>(h_t, 256, 1, tW2, NF,
                                                       tb2, emb, NF, 256, 1);

  // ---- feature MLP on transposed observation:
  //      obs_r[l, j] = observation[j*64 + l]  (strided A view) ----
  gemm_wmma_f32<false><<<dim3(64, 4), w, 0, stream>>>(observation, 1, 64,
                                                      fW1, 1024, fb1, hf1,
                                                      1024, 4096, 0);
  ln_relu_rows<<<64, 256, 0, stream>>>(hf1, f1g, f1b, 1024);
  gemm_wmma_f32<true><<<dim3(32, 4), w, 0, stream>>>(hf1, 1024, 1, fW2, 512,
                                                     fb2, hf2, 512, 1024, 0);
  ln_relu_rows<<<64, 256, 0, stream>>>(hf2, f2g, f2b, 512);
  gemm_wmma_f32<true><<<dim3(4096, 4), w, 0, stream>>>(hf2, 512, 1, fW3, NF,
                                                       fb3, Xflat, NF, 512, 0);

  // ---- X0[n,f,l] = Xflat[l, n*F + f] ----
  {
    const long tot = (long)NF * LDIM;
    xflat_to_x<<<(unsigned)((tot + 255) / 256), 256, 0, stream>>>(Xflat, X0);
  }

  // ================= dilation stage 1: d=2, gamma=1, L0=64, m=32 ============
  {
    const long zn = 32L * NF;
    zero_f32<<<(unsigned)((zn + 255) / 256), 256, 0, stream>>>(Z, zn);
    for (int k = 0; k < 2; ++k)
      graph_conv_wmma<<<dim3(64, 32), w, 0, stream>>>(
          emb, Bmat, X0, Wf + (long)k * 4096, Wb + (long)k * 4096, bconv, Z,
          64, 2, k, 1);
    // out[0] = Z[m=31]  (contiguous (N,F) slice)
    copy_f32<<<(NF + 255) / 256, 256, 0, stream>>>(Z + 31L * NF, out, NF);
    // X1[n,f,m] = Z[m,n,f]
    const long tot = 32L * NF;
    z_to_x<<<(unsigned)((tot + 255) / 256), 256, 0, stream>>>(Z, X1, 32);
  }

  // ================= dilation stage 2: d=2, gamma=2, L0=32, m=16 ============
  {
    const long zn = 16L * NF;
    zero_f32<<<(unsigned)((zn + 255) / 256), 256, 0, stream>>>(Z, zn);
    for (int k = 0; k < 2; ++k)
      graph_conv_wmma<<<dim3(64, 16), w, 0, stream>>>(
          emb, Bmat, X1, Wf + (long)k * 4096, Wb + (long)k * 4096, bconv, Z,
          32, 2, k, 2);
    // out[1] = Z[m=15]
    copy_f32<<<(NF + 255) / 256, 256, 0, stream>>>(Z + 15L * NF, out + NF, NF);
  }
}